// SPAGCN_71648644431957
// MI455X (gfx1250) — compile-verified
//
#include <hip/hip_runtime.h>
#include <math.h>

// ---------------------------------------------------------------------------
// SPAGCN forward: 2-layer GCN + pairwise soft-assignment q on MI455X (gfx1250)
// N=10000, E=160000, IN_DIM=1000, HID=256, OUT=2, ALPHA=BETA=0.5
// ---------------------------------------------------------------------------

#define GN   10000
#define GE   160000
#define GK   1000
#define GH   256
#define GOUT 2

typedef __attribute__((ext_vector_type(2))) float v2f;
typedef __attribute__((ext_vector_type(8))) float v8f;

// ---------------- utility: zero a float buffer ----------------
__global__ void k_zero_f32(float* __restrict__ p, int n) {
    int i = blockIdx.x * blockDim.x + threadIdx.x;
    if (i < n) p[i] = 0.0f;
}

// ---------------- degree / normalization ----------------
__global__ void k_deg_init(float* __restrict__ dinv, int n) {
    int i = blockIdx.x * blockDim.x + threadIdx.x;
    if (i < n) dinv[i] = 1.0f;                  // self-loop contributes 1
}

__global__ void k_deg_count(const int* __restrict__ dst, float* __restrict__ dinv, int e) {
    int i = blockIdx.x * blockDim.x + threadIdx.x;
    if (i < e) atomicAdd(&dinv[dst[i]], 1.0f);
}

__global__ void k_deg_finalize(float* __restrict__ dinv, int n) {
    int i = blockIdx.x * blockDim.x + threadIdx.x;
    if (i < n) dinv[i] = rsqrtf(fmaxf(dinv[i], 1.0f));
}

// ---------------- layer-1 GEMM: xw = X(10000x1000) @ W1(1000x256), fp32 WMMA
// One wave per 16x16 output tile; 4 waves (128 threads) per block.
// Tiles: (GN/16)=625 rows x (GH/16)=16 cols = 10000 tiles = 2500 blocks.
// A 16x4 layout: lane<16 -> M=lane, K=k+0/k+1 ; lane>=16 -> K=k+2/k+3
// B 4x16 layout: lane%16 = N column, same K split.
// C/D: lane%16 = N, vgpr r -> row r (lanes<16) / r+8 (lanes>=16)
__global__ void __launch_bounds__(128)
k_gemm1_wmma(const float* __restrict__ X, const float* __restrict__ W,
             float* __restrict__ Y) {
    const int lane   = threadIdx.x & 31;
    const int waveId = threadIdx.x >> 5;
    const int tile   = blockIdx.x * 4 + waveId;          // 0..9999, wave-uniform
    if (tile >= (GN / 16) * (GH / 16)) return;           // uniform exit: EXEC stays all-1s

    const int tileM = tile / (GH / 16);
    const int tileN = tile % (GH / 16);

    const int hi    = lane >> 4;                          // 0 or 1
    const int lm    = lane & 15;
    const int row   = tileM * 16 + lm;                    // A row for this lane
    const int col   = tileN * 16 + lm;                    // B column for this lane
    const int koff  = hi * 2;                             // K sub-offset per half-wave

    const float* __restrict__ xrow = X + (size_t)row * GK;

    v8f acc0 = {};
    v8f acc1 = {};
    // K = 1000, step 8 (two chained 16x16x4 WMMAs on independent accumulators)
    for (int k = 0; k < GK; k += 8) {
        v2f a0, b0, a1, b1v;
        a0.x  = xrow[k + koff];
        a0.y  = xrow[k + koff + 1];
        b0.x  = W[(size_t)(k + koff) * GH + col];
        b0.y  = W[(size_t)(k + koff + 1) * GH + col];
        a1.x  = xrow[k + 4 + koff];
        a1.y  = xrow[k + 4 + koff + 1];
        b1v.x = W[(size_t)(k + 4 + koff) * GH + col];
        b1v.y = W[(size_t)(k + 4 + koff + 1) * GH + col];
        acc0 = __builtin_amdgcn_wmma_f32_16x16x4_f32(
            false, a0, false, b0, (short)0, acc0, false, false);
        acc1 = __builtin_amdgcn_wmma_f32_16x16x4_f32(
            false, a1, false, b1v, (short)0, acc1, false, false);
    }

    const int rowbase = tileM * 16 + hi * 8;
    #pragma unroll
    for (int r = 0; r < 8; ++r) {
        Y[(size_t)(rowbase + r) * GH + col] = acc0[r] + acc1[r];
    }
}

// ---------------- layer-1 aggregation (edges): h[dst] += xw[src]*norm ------
// One block per edge, one thread per channel (coalesced read + atomic).
__global__ void __launch_bounds__(256)
k_agg1_edges(const int* __restrict__ src, const int* __restrict__ dst,
             const float* __restrict__ dinv, const float* __restrict__ xw,
             float* __restrict__ h) {
    const int e = blockIdx.x;
    const int c = threadIdx.x;
    const int s = src[e];
    const int d = dst[e];
    const float nrm = dinv[s] * dinv[d];
    atomicAdd(&h[(size_t)d * GH + c], xw[(size_t)s * GH + c] * nrm);
}

// self-loop + bias + ReLU (after all edge atomics complete)
__global__ void __launch_bounds__(256)
k_self1_bias_relu(const float* __restrict__ xw, const float* __restrict__ dinv,
                  const float* __restrict__ b1, float* __restrict__ h) {
    const int i = blockIdx.x;
    const int c = threadIdx.x;
    const float di2 = dinv[i] * dinv[i];
    float v = h[(size_t)i * GH + c] + xw[(size_t)i * GH + c] * di2 + b1[c];
    h[(size_t)i * GH + c] = v > 0.0f ? v : 0.0f;
}

// ---------------- layer-2 GEMM: hw2 = h(10000x256) @ W2(256x2) -------------
__global__ void k_gemm2(const float* __restrict__ h, const float* __restrict__ W2,
                        float* __restrict__ hw2) {
    int idx = blockIdx.x * blockDim.x + threadIdx.x;   // over N*2
    if (idx >= GN * GOUT) return;
    const int i = idx >> 1;
    const int o = idx & 1;
    float s = 0.0f;
    #pragma unroll 8
    for (int k = 0; k < GH; ++k)
        s = fmaf(h[(size_t)i * GH + k], W2[k * GOUT + o], s);
    hw2[idx] = s;
}

// ---------------- layer-2 aggregation (edges) ------------------------------
__global__ void k_agg2_edges(const int* __restrict__ src, const int* __restrict__ dst,
                             const float* __restrict__ dinv,
                             const float* __restrict__ hw2, float* __restrict__ emb) {
    int e = blockIdx.x * blockDim.x + threadIdx.x;
    if (e >= GE) return;
    const int s = src[e];
    const int d = dst[e];
    const float nrm = dinv[s] * dinv[d];
    atomicAdd(&emb[d * GOUT + 0], hw2[s * GOUT + 0] * nrm);
    atomicAdd(&emb[d * GOUT + 1], hw2[s * GOUT + 1] * nrm);
}

// self-loop + bias, then per-node squared norm
__global__ void k_self2_bias_sq(const float* __restrict__ hw2, const float* __restrict__ dinv,
                                const float* __restrict__ b2, float* __restrict__ emb,
                                float* __restrict__ sq) {
    int i = blockIdx.x * blockDim.x + threadIdx.x;
    if (i >= GN) return;
    const float di2 = dinv[i] * dinv[i];
    float e0 = emb[i * 2 + 0] + hw2[i * 2 + 0] * di2 + b2[0];
    float e1 = emb[i * 2 + 1] + hw2[i * 2 + 1] * di2 + b2[1];
    emb[i * 2 + 0] = e0;
    emb[i * 2 + 1] = e1;
    sq[i] = e0 * e0 + e1 * e1;
}

// ---------------- q matrix: q[i,j] = 1/(1 + 0.5*dist(i,j)) -----------------
// 400 MB streamed store — the bandwidth floor (~17us at 23.3 TB/s).
__global__ void __launch_bounds__(256)
k_qmat(const float* __restrict__ emb, const float* __restrict__ sq,
       float* __restrict__ q) {
    const int j = blockIdx.x * blockDim.x + threadIdx.x;  // fast (coalesced) dim
    const int i = blockIdx.y;
    if (j >= GN) return;
    const float e0 = emb[i * 2 + 0];
    const float e1 = emb[i * 2 + 1];
    const float f0 = emb[j * 2 + 0];
    const float f1 = emb[j * 2 + 1];
    float d2 = sq[i] + sq[j] - 2.0f * (e0 * f0 + e1 * f1);
    d2 = fmaxf(d2, 0.0f);
    const float dist = sqrtf(d2);                          // dist^(2*0.5) == dist
    q[(size_t)i * GN + j] = 1.0f / (1.0f + 0.5f * dist);   // ALPHA = 0.5
}

// ---------------------------------------------------------------------------
extern "C" void kernel_launch(void* const* d_in, const int* in_sizes, int n_in,
                              void* d_out, int out_size, void* d_ws, size_t ws_size,
                              hipStream_t stream) {
    const float* X   = (const float*)d_in[0];   // [N, IN_DIM]
    const int*   ei  = (const int*)  d_in[1];   // [2, E] flattened
    const float* W1  = (const float*)d_in[2];   // [IN_DIM, HID]
    const float* b1  = (const float*)d_in[3];   // [HID]
    const float* W2  = (const float*)d_in[4];   // [HID, OUT]
    const float* b2  = (const float*)d_in[5];   // [OUT]

    const int E = in_sizes[1] / 2;              // 160000
    const int* src = ei;
    const int* dst = ei + E;

    // workspace layout (floats)
    char* ws = (char*)d_ws;
    float* xw   = (float*)ws;                                   // N*HID
    float* h    = (float*)(ws + (size_t)GN * GH * 4);           // N*HID
    float* dinv = (float*)(ws + (size_t)2 * GN * GH * 4);       // N
    float* hw2  = dinv + GN;                                    // N*2
    float* sq   = hw2 + GN * GOUT;                              // N

    float* emb = (float*)d_out;                 // first N*2 floats of output
    float* q   = emb + (size_t)GN * GOUT;       // N*N floats

    const int T = 256;

    // 1) degrees -> dinv = rsqrt(max(deg,1))
    k_deg_init    <<<(GN + T - 1) / T, T, 0, stream>>>(dinv, GN);
    k_deg_count   <<<(E  + T - 1) / T, T, 0, stream>>>(dst, dinv, E);
    k_deg_finalize<<<(GN + T - 1) / T, T, 0, stream>>>(dinv, GN);

    // 2) xw = X @ W1  (fp32 WMMA, 1 wave per 16x16 tile, 4 waves/block)
    k_gemm1_wmma<<<(GN / 16) * (GH / 16) / 4, 128, 0, stream>>>(X, W1, xw);

    // 3) h = relu(scatter(norm * xw) + self + b1)
    k_zero_f32<<<((GN * GH) + T - 1) / T, T, 0, stream>>>(h, GN * GH);
    k_agg1_edges<<<E, GH, 0, stream>>>(src, dst, dinv, xw, h);
    k_self1_bias_relu<<<GN, GH, 0, stream>>>(xw, dinv, b1, h);

    // 4) hw2 = h @ W2
    k_gemm2<<<((GN * GOUT) + T - 1) / T, T, 0, stream>>>(h, W2, hw2);

    // 5) emb = scatter(norm * hw2) + self + b2 ; sq = |emb|^2
    k_zero_f32<<<((GN * GOUT) + T - 1) / T, T, 0, stream>>>(emb, GN * GOUT);
    k_agg2_edges<<<(E + T - 1) / T, T, 0, stream>>>(src, dst, dinv, hw2, emb);
    k_self2_bias_sq<<<(GN + T - 1) / T, T, 0, stream>>>(hw2, dinv, b2, emb, sq);

    // 6) q = 1/(1 + 0.5*dist)
    dim3 qgrid((GN + T - 1) / T, GN);
    k_qmat<<<qgrid, T, 0, stream>>>(emb, sq, q);
}